// MPGNNEncoder_77369540870568
// MI455X (gfx1250) — compile-verified
//
#include <hip/hip_runtime.h>

#define N_NODES 50000
#define N_EDGES 800000
#define DIM     128

typedef __attribute__((ext_vector_type(2))) float v2f;
typedef __attribute__((ext_vector_type(8))) float v8f;

// ---------------------------------------------------------------------------
// Degree / normalization kernels
// ---------------------------------------------------------------------------
__global__ void init_deg_kernel(float* __restrict__ deg, int n) {
  int i = blockIdx.x * blockDim.x + threadIdx.x;
  if (i < n) deg[i] = 1.0f;  // self-loop contributes 1 to every node's degree
}

__global__ void count_deg_kernel(const int* __restrict__ dst,
                                 float* __restrict__ deg, int e) {
  int i = blockIdx.x * blockDim.x + threadIdx.x;
  if (i < e) unsafeAtomicAdd(&deg[dst[i]], 1.0f);
}

__global__ void rsqrt_kernel(float* __restrict__ deg, int n) {
  int i = blockIdx.x * blockDim.x + threadIdx.x;
  if (i < n) deg[i] = rsqrtf(deg[i]);  // deg >= 1 always (self-loop)
}

// ---------------------------------------------------------------------------
// WMMA GEMM: H = X @ W (fp32, V_WMMA_F32_16X16X4_F32), fused self-loop init:
// AGG[r] = dinv[r]^2 * H[r]
// Block = 256 threads (8 wave32). Block b computes rows [16b, 16b+16).
// Wave w computes the 16x16 output tile at columns [16w, 16w+16).
// ---------------------------------------------------------------------------
__launch_bounds__(256)
__global__ void gcn_gemm_wmma_kernel(const float* __restrict__ X,
                                     const float* __restrict__ W,
                                     const float* __restrict__ dinv,
                                     float* __restrict__ H,
                                     float* __restrict__ AGG) {
  __shared__ float sW[DIM * DIM];  // 64 KB: full 128x128 weight tile
  __shared__ float sA[16 * DIM];   // 8 KB: 16-row A stripe

  const int tid  = threadIdx.x;
  const int row0 = blockIdx.x * 16;

  // Stage W and the A stripe into LDS (float4 vectorized).
  {
    const float4* Wv  = (const float4*)W;
    float4*       sWv = (float4*)sW;
    #pragma unroll
    for (int i = tid; i < (DIM * DIM) / 4; i += 256) sWv[i] = Wv[i];

    const float4* Xv  = (const float4*)(X + (size_t)row0 * DIM);
    float4*       sAv = (float4*)sA;
    for (int i = tid; i < (16 * DIM) / 4; i += 256) sAv[i] = Xv[i];
  }
  __syncthreads();

  const int wave = tid >> 5;
  const int lane = tid & 31;
  const int half = lane >> 4;   // 0: lanes 0-15, 1: lanes 16-31
  const int l16  = lane & 15;
  const int col0 = wave * 16;

  v8f acc = {};

  // A 16x4 layout (ISA 7.12.2): lanes 0-15 hold K={k0,k0+1}, lanes 16-31 hold
  // K={k0+2,k0+3} for row M=lane%16 -> one contiguous float2 per lane.
  // B 4x16: VGPR0 = rows K={k0 | k0+2} striped over lanes, VGPR1 = K+1.
  #pragma unroll 8
  for (int k0 = 0; k0 < DIM; k0 += 4) {
    v2f a = *(const v2f*)&sA[l16 * DIM + k0 + 2 * half];
    v2f b;
    b.x = sW[(k0 + 2 * half) * DIM + col0 + l16];
    b.y = sW[(k0 + 2 * half + 1) * DIM + col0 + l16];
    acc = __builtin_amdgcn_wmma_f32_16x16x4_f32(
        /*neg_a=*/false, a, /*neg_b=*/false, b,
        /*c_mod=*/(short)0, acc, /*reuse_a=*/false, /*reuse_b=*/false);
  }

  // C/D layout: lanes 0-15 -> M = j, lanes 16-31 -> M = 8 + j (VGPR j).
  #pragma unroll
  for (int j = 0; j < 8; ++j) {
    int   r = row0 + half * 8 + j;
    float v = acc[j];
    float w = dinv[r];
    H[(size_t)r * DIM + col0 + l16]   = v;
    AGG[(size_t)r * DIM + col0 + l16] = (w * w) * v;  // self-loop message
  }
}

// ---------------------------------------------------------------------------
// Edge scatter: AGG[dst] += dinv[src]*dinv[dst] * H[src]
// One wave per edge; each lane handles 4 channels (float4 gather + 4 atomics).
// ---------------------------------------------------------------------------
__launch_bounds__(256)
__global__ void edge_scatter_kernel(const int* __restrict__ src,
                                    const int* __restrict__ dst,
                                    const float* __restrict__ dinv,
                                    const float* __restrict__ H,
                                    float* __restrict__ AGG) {
  int e = blockIdx.x * 8 + (threadIdx.x >> 5);
  if (e >= N_EDGES) return;
  const int lane = threadIdx.x & 31;

  const int   s = src[e];
  const int   d = dst[e];
  const float w = dinv[s] * dinv[d];

  const float4 hv = *(const float4*)&H[(size_t)s * DIM + lane * 4];
  float* base = &AGG[(size_t)d * DIM + lane * 4];
  unsafeAtomicAdd(base + 0, w * hv.x);
  unsafeAtomicAdd(base + 1, w * hv.y);
  unsafeAtomicAdd(base + 2, w * hv.z);
  unsafeAtomicAdd(base + 3, w * hv.w);
}

// ---------------------------------------------------------------------------
// out = elu(in + bias)   (alpha = 1; elementwise, may run in place)
// ---------------------------------------------------------------------------
__global__ void bias_elu_kernel(const float* __restrict__ in,
                                const float* __restrict__ bias,
                                float* __restrict__ out, int total) {
  int i = blockIdx.x * blockDim.x + threadIdx.x;
  if (i < total) {
    float v = in[i] + bias[i & (DIM - 1)];
    out[i] = (v > 0.0f) ? v : (__expf(v) - 1.0f);
  }
}

// ---------------------------------------------------------------------------
extern "C" void kernel_launch(void* const* d_in, const int* in_sizes, int n_in,
                              void* d_out, int out_size, void* d_ws,
                              size_t ws_size, hipStream_t stream) {
  const float* x    = (const float*)d_in[0];
  const int*   eidx = (const int*)d_in[1];   // [2, E] row-major
  const float* W1   = (const float*)d_in[2];
  const float* b1   = (const float*)d_in[3];
  const float* W2   = (const float*)d_in[4];
  const float* b2   = (const float*)d_in[5];
  float*       out  = (float*)d_out;

  const int* src = eidx;            // edge_index[0]
  const int* dst = eidx + N_EDGES;  // edge_index[1]

  float* dinv = (float*)d_ws;                         // N floats
  float* bufA = dinv + N_NODES;                       // N*D floats
  float* bufB = bufA + (size_t)N_NODES * DIM;         // N*D floats

  const int total = N_NODES * DIM;

  // Normalization coefficients.
  init_deg_kernel<<<(N_NODES + 255) / 256, 256, 0, stream>>>(dinv, N_NODES);
  count_deg_kernel<<<(N_EDGES + 255) / 256, 256, 0, stream>>>(dst, dinv, N_EDGES);
  rsqrt_kernel<<<(N_NODES + 255) / 256, 256, 0, stream>>>(dinv, N_NODES);

  // Layer 1: H1 = x@W1 -> bufA ; agg init -> bufB ; scatter ; elu -> bufA
  gcn_gemm_wmma_kernel<<<N_NODES / 16, 256, 0, stream>>>(x, W1, dinv, bufA, bufB);
  edge_scatter_kernel<<<N_EDGES / 8, 256, 0, stream>>>(src, dst, dinv, bufA, bufB);
  bias_elu_kernel<<<(total + 255) / 256, 256, 0, stream>>>(bufB, b1, bufA, total);

  // Layer 2: H2 = act1@W2 -> bufB ; agg init -> d_out ; scatter ; elu in place
  gcn_gemm_wmma_kernel<<<N_NODES / 16, 256, 0, stream>>>(bufA, W2, dinv, bufB, out);
  edge_scatter_kernel<<<N_EDGES / 8, 256, 0, stream>>>(src, dst, dinv, bufB, out);
  bias_elu_kernel<<<(total + 255) / 256, 256, 0, stream>>>(out, b2, out, total);
}